// GGC_30374008717357
// MI455X (gfx1250) — compile-verified
//
#include <hip/hip_runtime.h>
#include <hip/hip_bf16.h>
#include <math.h>

// ---------------------------------------------------------------------------
// Gated GNN (2x GatedGraphConv-style layers + log_softmax) for gfx1250.
// Dense GEMMs via V_WMMA_F32_16X16X4_F32 (fp32 matrix core path).
// Edge gather/scatter via float4 loads + global f32 atomics (L2-resident agg).
// ---------------------------------------------------------------------------

typedef __attribute__((ext_vector_type(2))) float v2f;
typedef __attribute__((ext_vector_type(8))) float v8f;

#define NN 100000   // nodes
#define NE 2000000  // edges
#define NCLS 40

static __device__ __forceinline__ v8f wmma4(v2f a, v2f b, v8f c) {
  // D = A(16x4,f32) * B(4x16,f32) + C(16x16,f32)
  return __builtin_amdgcn_wmma_f32_16x16x4_f32(false, a, false, b, (short)0, c,
                                               false, false);
}

static __device__ __forceinline__ v8f zero8() {
  v8f z;
#pragma unroll
  for (int i = 0; i < 8; ++i) z[i] = 0.0f;
  return z;
}

// A fragment (16x4) from row-major src[.. x ld], rows row0..row0+15, K block k.
// lane l: m = l&15, holds K = k + 2*(l>>4) + {0,1}  (ISA 7.12.2, 32-bit A 16x4)
static __device__ __forceinline__ v2f loadA_rm(const float* __restrict__ src,
                                               int ld, int row0, int k, int lane) {
  int m = lane & 15, kh = (lane >> 4) * 2;
  const float* p = src + (size_t)(row0 + m) * ld + (k + kh);
  v2f a; a.x = p[0]; a.y = p[1];
  return a;
}

// B fragment (4x16) from row-major W[K x ld]; columns n0..n0+15, guarded n<ncols
static __device__ __forceinline__ v2f loadB_rm(const float* __restrict__ W,
                                               int ld, int k, int n0, int ncols,
                                               int lane) {
  int n = n0 + (lane & 15), kh = (lane >> 4) * 2;
  v2f b;
  if (n < ncols) {
    b.x = W[(size_t)(k + kh) * ld + n];
    b.y = W[(size_t)(k + kh + 1) * ld + n];
  } else { b.x = 0.0f; b.y = 0.0f; }
  return b;
}

// B fragment from a transposed weight: B[k][n] = W[n*ld + k]  (W is [ncols x ld])
static __device__ __forceinline__ v2f loadB_tr(const float* __restrict__ W,
                                               int ld, int k, int n0, int ncols,
                                               int lane) {
  int n = n0 + (lane & 15), kh = (lane >> 4) * 2;
  v2f b;
  if (n < ncols) {
    const float* p = W + (size_t)n * ld + (k + kh);
    b.x = p[0]; b.y = p[1];
  } else { b.x = 0.0f; b.y = 0.0f; }
  return b;
}

// ---------------------------------------------------------------------------
// prep: H = Xin(16xK1) @ Wp(K1 x F);  M = H @ Wm(F x F).  One 16-row tile/wave.
// ---------------------------------------------------------------------------
template <int K1, int F, int FT>
__global__ void __launch_bounds__(128) prep_kernel(
    const float* __restrict__ xin, const float* __restrict__ Wp,
    const float* __restrict__ Wm, float* __restrict__ hout,
    float* __restrict__ mout) {
  __shared__ float sh[4][16 * F];
  const int lane = threadIdx.x & 31;
  const int wave = threadIdx.x >> 5;
  const int row0 = (blockIdx.x * 4 + wave) * 16;
  if (row0 >= NN) return;  // wave-uniform exit; active waves keep EXEC = all-1

  v8f cH[FT];
#pragma unroll
  for (int t = 0; t < FT; ++t) cH[t] = zero8();

#pragma unroll
  for (int k = 0; k < K1; k += 4) {
    v2f a = loadA_rm(xin, K1, row0, k, lane);
#pragma unroll
    for (int t = 0; t < FT; ++t) {
      v2f b = loadB_rm(Wp, F, k, t * 16, F, lane);
      cH[t] = wmma4(a, b, cH[t]);
    }
  }

  // Store H tile: global + LDS (C layout: VGPR j -> row j + 8*(lane>=16))
  const int rbase = (lane >> 4) * 8;
  const int cl = lane & 15;
#pragma unroll
  for (int t = 0; t < FT; ++t) {
    int col = t * 16 + cl;
    if (col < F) {
#pragma unroll
      for (int j = 0; j < 8; ++j) {
        int r = rbase + j;
        float v = cH[t][j];
        hout[(size_t)(row0 + r) * F + col] = v;
        sh[wave][r * F + col] = v;
      }
    }
  }
  asm volatile("s_wait_dscnt 0" ::: "memory");  // intra-wave LDS exchange

  v8f cM[FT];
#pragma unroll
  for (int t = 0; t < FT; ++t) cM[t] = zero8();
#pragma unroll
  for (int k = 0; k < F; k += 4) {
    v2f a = loadA_rm(&sh[wave][0], F, 0, k, lane);
#pragma unroll
    for (int t = 0; t < FT; ++t) {
      v2f b = loadB_rm(Wm, F, k, t * 16, F, lane);
      cM[t] = wmma4(a, b, cM[t]);
    }
  }
#pragma unroll
  for (int t = 0; t < FT; ++t) {
    int col = t * 16 + cl;
    if (col < F) {
#pragma unroll
      for (int j = 0; j < 8; ++j)
        mout[(size_t)(row0 + rbase + j) * F + col] = cM[t][j];
    }
  }
}

// ---------------------------------------------------------------------------
// edge scatter: agg[dst] += msg[src] * w   (float4 gather, f32 atomics to L2)
// ---------------------------------------------------------------------------
template <int F>
__global__ void __launch_bounds__(256) edge_kernel(
    const float* __restrict__ msg, const int* __restrict__ ei,
    const float* __restrict__ ew, float* __restrict__ agg) {
  const int VPE = F / 4;
  int tid = blockIdx.x * blockDim.x + threadIdx.x;
  if (tid >= NE * VPE) return;
  int e = tid / VPE, f4 = tid - e * VPE;
  int src = ei[e];
  int dst = ei[NE + e];
  float w = ew[e];
  float4 v = *((const float4*)(msg + (size_t)src * F) + f4);
  float* ap = agg + (size_t)dst * F + f4 * 4;
  atomicAdd(ap + 0, v.x * w);
  atomicAdd(ap + 1, v.y * w);
  atomicAdd(ap + 2, v.z * w);
  atomicAdd(ap + 3, v.w * w);
}

// ---------------------------------------------------------------------------
// GRU cell: gi = agg@Wih^T + bih ; gh = h@Whh^T + bhh ; gates r,z,n; opt ReLU.
// HT = ceil(3H/16) tiles; gate columns recombined via LDS (H=40 misaligns).
// ---------------------------------------------------------------------------
template <int H, int HT, bool RELU>
__global__ void __launch_bounds__(128) gru_kernel(
    const float* __restrict__ agg, const float* __restrict__ hmat,
    const float* __restrict__ wih, const float* __restrict__ whh,
    const float* __restrict__ bih, const float* __restrict__ bhh,
    float* __restrict__ out) {
  __shared__ float sgi[4][16 * 3 * H];
  __shared__ float sgh[4][16 * 3 * H];
  const int lane = threadIdx.x & 31;
  const int wave = threadIdx.x >> 5;
  const int row0 = (blockIdx.x * 4 + wave) * 16;
  if (row0 >= NN) return;

  v8f cgi[HT], cgh[HT];
#pragma unroll
  for (int t = 0; t < HT; ++t) { cgi[t] = zero8(); cgh[t] = zero8(); }

#pragma unroll
  for (int k = 0; k < H; k += 4) {
    v2f aA = loadA_rm(agg, H, row0, k, lane);
    v2f aH = loadA_rm(hmat, H, row0, k, lane);
#pragma unroll
    for (int t = 0; t < HT; ++t) {
      v2f bI = loadB_tr(wih, H, k, t * 16, 3 * H, lane);
      v2f bH = loadB_tr(whh, H, k, t * 16, 3 * H, lane);
      cgi[t] = wmma4(aA, bI, cgi[t]);
      cgh[t] = wmma4(aH, bH, cgh[t]);
    }
  }

  const int rbase = (lane >> 4) * 8;
  const int cl = lane & 15;
#pragma unroll
  for (int t = 0; t < HT; ++t) {
    int col = t * 16 + cl;
    if (col < 3 * H) {
#pragma unroll
      for (int j = 0; j < 8; ++j) {
        int r = rbase + j;
        sgi[wave][r * 3 * H + col] = cgi[t][j];
        sgh[wave][r * 3 * H + col] = cgh[t][j];
      }
    }
  }
  asm volatile("s_wait_dscnt 0" ::: "memory");

  for (int idx = lane; idx < 16 * H; idx += 32) {
    int m = idx / H, c = idx - m * H;
    const float* gi = &sgi[wave][m * 3 * H];
    const float* gh = &sgh[wave][m * 3 * H];
    float ir = gi[c] + bih[c];
    float iz = gi[H + c] + bih[H + c];
    float in_ = gi[2 * H + c] + bih[2 * H + c];
    float hr = gh[c] + bhh[c];
    float hz = gh[H + c] + bhh[H + c];
    float hn = gh[2 * H + c] + bhh[2 * H + c];
    float r = 1.0f / (1.0f + expf(-(ir + hr)));
    float z = 1.0f / (1.0f + expf(-(iz + hz)));
    float n = tanhf(in_ + r * hn);
    float hv = hmat[(size_t)(row0 + m) * H + c];
    float o = (1.0f - z) * n + z * hv;
    if (RELU) o = fmaxf(o, 0.0f);
    out[(size_t)(row0 + m) * H + c] = o;
  }
}

// ---------------------------------------------------------------------------
// log_softmax over NCLS classes, in place on d_out.
// ---------------------------------------------------------------------------
__global__ void __launch_bounds__(256) lsm_kernel(float* __restrict__ out) {
  int n = blockIdx.x * blockDim.x + threadIdx.x;
  if (n >= NN) return;
  float* row = out + (size_t)n * NCLS;
  float mx = -INFINITY;
#pragma unroll
  for (int c = 0; c < NCLS; ++c) mx = fmaxf(mx, row[c]);
  float s = 0.0f;
#pragma unroll
  for (int c = 0; c < NCLS; ++c) s += expf(row[c] - mx);
  float lse = mx + logf(s);
#pragma unroll
  for (int c = 0; c < NCLS; ++c) row[c] = row[c] - lse;
}

// ---------------------------------------------------------------------------
extern "C" void kernel_launch(void* const* d_in, const int* in_sizes, int n_in,
                              void* d_out, int out_size, void* d_ws,
                              size_t ws_size, hipStream_t stream) {
  (void)in_sizes; (void)n_in; (void)out_size; (void)ws_size;
  const float* x   = (const float*)d_in[0];
  const int*   ei  = (const int*)d_in[1];
  const float* ew  = (const float*)d_in[2];
  const float* W1p = (const float*)d_in[3];
  const float* W1m = (const float*)d_in[4];
  const float* g1wih = (const float*)d_in[5];
  const float* g1whh = (const float*)d_in[6];
  const float* g1bih = (const float*)d_in[7];
  const float* g1bhh = (const float*)d_in[8];
  const float* W2p = (const float*)d_in[9];
  const float* W2m = (const float*)d_in[10];
  const float* g2wih = (const float*)d_in[11];
  const float* g2whh = (const float*)d_in[12];
  const float* g2bih = (const float*)d_in[13];
  const float* g2bhh = (const float*)d_in[14];
  float* out = (float*)d_out;
  float* ws  = (float*)d_ws;

  const size_t N = NN;
  // Workspace layout (floats), lifetime-safe aliasing; total 168*N ~= 67 MB:
  float* h1   = ws;             // [0,   32N)  layer-1 h
  float* m1   = ws + 32 * N;    // [32N, 64N)  layer-1 msg
  float* agg1 = ws + 64 * N;    // [64N, 96N)
  float* h1r  = ws + 96 * N;    // [96N,128N)  relu(gru1)
  float* h2   = ws;             // [0,   40N)  reuses h1/m1 (dead after gru1)
  float* m2   = ws + 40 * N;    // [40N, 80N)  reuses m1/agg1 (dead)
  float* agg2 = ws + 128 * N;   // [128N,168N)

  hipMemsetAsync(agg1, 0, N * 32 * sizeof(float), stream);
  hipMemsetAsync(agg2, 0, N * 40 * sizeof(float), stream);

  const int tiles = NN / 16;                 // 6250 exactly
  const int blocks = (tiles + 3) / 4;        // 4 waves (tiles) per block

  // Layer 1
  prep_kernel<32, 32, 2><<<blocks, 128, 0, stream>>>(x, W1p, W1m, h1, m1);
  edge_kernel<32><<<(NE * 8 + 255) / 256, 256, 0, stream>>>(m1, ei, ew, agg1);
  gru_kernel<32, 6, true><<<blocks, 128, 0, stream>>>(
      agg1, h1, g1wih, g1whh, g1bih, g1bhh, h1r);

  // Layer 2
  prep_kernel<32, 40, 3><<<blocks, 128, 0, stream>>>(h1r, W2p, W2m, h2, m2);
  edge_kernel<40><<<(NE * 10 + 255) / 256, 256, 0, stream>>>(m2, ei, ew, agg2);
  gru_kernel<40, 8, false><<<blocks, 128, 0, stream>>>(
      agg2, h2, g2wih, g2whh, g2bih, g2bhh, out);

  lsm_kernel<<<(NN + 255) / 256, 256, 0, stream>>>(out);
}